// GAT_32908039422447
// MI455X (gfx1250) — compile-verified
//
#include <hip/hip_runtime.h>

#define WG 256

typedef float v2f __attribute__((ext_vector_type(2)));
typedef float v8f __attribute__((ext_vector_type(8)));

// ---------- order-preserving float<->uint encoding for atomic max ----------
__device__ __forceinline__ unsigned enc_f32(float f) {
    unsigned u = __float_as_uint(f);
    return (u & 0x80000000u) ? ~u : (u | 0x80000000u);
}
__device__ __forceinline__ float dec_f32(unsigned e) {
    unsigned b = (e & 0x80000000u) ? (e & 0x7FFFFFFFu) : ~e;
    return __uint_as_float(b);
}
#define ENC_NEG_INF 0x007FFFFFu   // enc_f32(-inf)

__device__ __forceinline__ void edge_sd(const long long* __restrict__ ei,
                                        long long E, long long i, int& s, int& d) {
    if (i < E) { s = (int)ei[i]; d = (int)ei[E + i]; }
    else       { s = d = (int)(i - E); }          // self loop
}

// ---------------------------- init kernels ---------------------------------
__global__ void zero_f32(float* __restrict__ p, long long n) {
    long long i = (long long)blockIdx.x * blockDim.x + threadIdx.x;
    long long st = (long long)gridDim.x * blockDim.x;
    for (; i < n; i += st) p[i] = 0.0f;
}
__global__ void fill_u32(unsigned* __restrict__ p, unsigned v, long long n) {
    long long i = (long long)blockIdx.x * blockDim.x + threadIdx.x;
    long long st = (long long)gridDim.x * blockDim.x;
    for (; i < n; i += st) p[i] = v;
}

// --------------------- layer-1 GEMM: h1 = x @ W1 (WMMA f32) ----------------
// x: [N,128] row-major, W: [128,64] row-major, h: [N,64]
// one wave computes one 16x16 tile; K-loop of 32 x V_WMMA_F32_16X16X4_F32
__global__ void __launch_bounds__(128) gemm1_wmma(const float* __restrict__ x,
                                                  const float* __restrict__ W,
                                                  float* __restrict__ h, int N) {
    const int lane = threadIdx.x & 31;
    const int wave = threadIdx.x >> 5;
    const int tile = blockIdx.x * 4 + wave;   // mtiles*4 tiles total
    const int mt = tile >> 2;                 // 16-row tile index
    const int nt = tile & 3;                  // 16-col tile index (64/16)
    const int m0 = mt * 16;
    const int n0 = nt * 16;
    const int half = lane >> 4;               // 0: K+{0,1} | 1: K+{2,3}
    const int l15  = lane & 15;

    int mrow = m0 + l15; if (mrow > N - 1) mrow = N - 1;   // clamp (no divergence)
    const float* __restrict__ xrow = x + (size_t)mrow * 128;

    v8f acc = {};
    for (int k = 0; k < 128; k += 4) {
        const int ka = k + half * 2;
        v2f a, b;
        a.x = xrow[ka];
        a.y = xrow[ka + 1];
        b.x = W[(size_t)ka * 64 + n0 + l15];
        b.y = W[(size_t)(ka + 1) * 64 + n0 + l15];
        acc = __builtin_amdgcn_wmma_f32_16x16x4_f32(
                  false, a, false, b, (short)0, acc, false, false);
    }
    // C/D layout: VGPR r -> row (half*8 + r), lanes l15 -> col n0+l15
    if (m0 + 16 <= N) {
        // fast path: wave-uniform, no per-row guard -> 8 stores, immediate offsets
        float* __restrict__ base = h + (size_t)(m0 + half * 8) * 64 + n0 + l15;
        #pragma unroll
        for (int r = 0; r < 8; ++r) base[(size_t)r * 64] = acc[r];
    } else {
        #pragma unroll
        for (int r = 0; r < 8; ++r) {
            int row = m0 + half * 8 + r;
            if (row < N) h[(size_t)row * 64 + n0 + l15] = acc[r];
        }
    }
}

// -------------------- per-node attention logits (layer 1) ------------------
__global__ void alpha_kernel(const float* __restrict__ h,
                             const float* __restrict__ a_s,
                             const float* __restrict__ a_d,
                             float* __restrict__ as, float* __restrict__ ad,
                             int N, int F) {
    int i = blockIdx.x * blockDim.x + threadIdx.x;
    if (i >= N) return;
    const float* row = h + (size_t)i * F;
    float s = 0.f, d = 0.f;
    for (int f = 0; f < F; ++f) { float v = row[f]; s += v * a_s[f]; d += v * a_d[f]; }
    as[i] = s; ad[i] = d;
}

// ----------- layer-2 projection g = h2@W2 (64->2) + logits -----------------
__global__ void proj2_kernel(const float* __restrict__ h2,
                             const float* __restrict__ W2,
                             const float* __restrict__ a_s,
                             const float* __restrict__ a_d,
                             float* __restrict__ g,
                             float* __restrict__ as, float* __restrict__ ad, int N) {
    int i = blockIdx.x * blockDim.x + threadIdx.x;
    if (i >= N) return;
    const float* row = h2 + (size_t)i * 64;
    float g0 = 0.f, g1 = 0.f;
    for (int f = 0; f < 64; ++f) {
        float v = row[f];
        g0 += v * W2[2 * f + 0];
        g1 += v * W2[2 * f + 1];
    }
    g[2 * i + 0] = g0; g[2 * i + 1] = g1;
    as[i] = g0 * a_s[0] + g1 * a_s[1];
    ad[i] = g0 * a_d[0] + g1 * a_d[1];
}

// ------------------------------ edge kernels -------------------------------
__global__ void edge_max_kernel(const long long* __restrict__ ei, long long E, int N,
                                const float* __restrict__ as,
                                const float* __restrict__ ad,
                                unsigned* __restrict__ menc) {
    long long i = (long long)blockIdx.x * blockDim.x + threadIdx.x;
    long long ET = E + N;
    if (i >= ET) return;
    int s, d; edge_sd(ei, E, i, s, d);
    float e = as[s] + ad[d];
    e = e > 0.f ? e : 0.2f * e;                     // leaky_relu(0.2)
    atomicMax(&menc[d], enc_f32(e));
}

__global__ void edge_p_kernel(const long long* __restrict__ ei, long long E, int N,
                              const float* __restrict__ as,
                              const float* __restrict__ ad,
                              const unsigned* __restrict__ menc,
                              float* __restrict__ p, float* __restrict__ denom) {
    long long i = (long long)blockIdx.x * blockDim.x + threadIdx.x;
    long long ET = E + N;
    if (i >= ET) return;
    int s, d; edge_sd(ei, E, i, s, d);
    float e = as[s] + ad[d];
    e = e > 0.f ? e : 0.2f * e;
    float m = dec_f32(menc[d]);
    if (!isfinite(m)) m = 0.f;                      // nodes w/o in-edges
    float pe = __expf(e - m);
    p[i] = pe;
    atomicAdd(&denom[d], pe);
}

// layer-1 aggregation: 4 threads per edge, 16 features each (float4 gathers)
__global__ void edge_agg64_kernel(const long long* __restrict__ ei, long long E, int N,
                                  const float* __restrict__ p,
                                  const float* __restrict__ denom,
                                  const float* __restrict__ h,
                                  float* __restrict__ agg) {
    long long t = (long long)blockIdx.x * blockDim.x + threadIdx.x;
    long long ET = E + N;
    if (t >= ET * 4) return;
    long long i = t >> 2;
    int c = (int)(t & 3);
    int s, d; edge_sd(ei, E, i, s, d);
    float alpha = p[i] / (denom[d] + 1e-16f);
    const float* hs = h + (size_t)s * 64 + c * 16;
    float* ag = agg + (size_t)d * 64 + c * 16;
    __builtin_prefetch(hs, 0, 0);                   // global_prefetch_b8
    const float4* hs4 = reinterpret_cast<const float4*>(hs);
    #pragma unroll
    for (int q = 0; q < 4; ++q) {
        float4 v = hs4[q];
        atomicAdd(&ag[4 * q + 0], alpha * v.x);
        atomicAdd(&ag[4 * q + 1], alpha * v.y);
        atomicAdd(&ag[4 * q + 2], alpha * v.z);
        atomicAdd(&ag[4 * q + 3], alpha * v.w);
    }
}

// layer-2 aggregation: 1 thread per edge, 2 features
__global__ void edge_agg2_kernel(const long long* __restrict__ ei, long long E, int N,
                                 const float* __restrict__ p,
                                 const float* __restrict__ denom,
                                 const float* __restrict__ g,
                                 float* __restrict__ agg) {
    long long i = (long long)blockIdx.x * blockDim.x + threadIdx.x;
    long long ET = E + N;
    if (i >= ET) return;
    int s, d; edge_sd(ei, E, i, s, d);
    float alpha = p[i] / (denom[d] + 1e-16f);
    atomicAdd(&agg[2 * d + 0], alpha * g[2 * s + 0]);
    atomicAdd(&agg[2 * d + 1], alpha * g[2 * s + 1]);
}

// ---------------------------- epilogues ------------------------------------
// F is a power of two (64 or 2) -> use mask instead of % to avoid int division
__global__ void bias_relu_kernel(float* __restrict__ a, const float* __restrict__ b,
                                 long long n, int Fmask) {
    long long i = (long long)blockIdx.x * blockDim.x + threadIdx.x;
    if (i >= n) return;
    float v = a[i] + b[(int)i & Fmask];
    a[i] = v > 0.f ? v : 0.f;
}
__global__ void bias_out_kernel(const float* __restrict__ a, const float* __restrict__ b,
                                float* __restrict__ out, long long n, int Fmask) {
    long long i = (long long)blockIdx.x * blockDim.x + threadIdx.x;
    if (i >= n) return;
    out[i] = a[i] + b[(int)i & Fmask];
}

// ---------------------------------------------------------------------------
extern "C" void kernel_launch(void* const* d_in, const int* in_sizes, int n_in,
                              void* d_out, int out_size, void* d_ws, size_t ws_size,
                              hipStream_t stream) {
    (void)n_in; (void)out_size; (void)ws_size;
    const float*     x    = (const float*)d_in[0];
    const long long* ei   = (const long long*)d_in[1];   // int64 [2,E]
    const float*     W1   = (const float*)d_in[2];
    const float*     a_s1 = (const float*)d_in[3];
    const float*     a_d1 = (const float*)d_in[4];
    const float*     b1   = (const float*)d_in[5];
    const float*     W2   = (const float*)d_in[6];
    const float*     a_s2 = (const float*)d_in[7];
    const float*     a_d2 = (const float*)d_in[8];
    const float*     b2   = (const float*)d_in[9];
    float* out = (float*)d_out;

    const int F_IN = 128, H = 64, C = 2;
    const int N = in_sizes[0] / F_IN;
    const long long E  = in_sizes[1] / 2;
    const long long ET = E + (long long)N;

    // workspace layout (floats)
    float* ws = (float*)d_ws;
    size_t o = 0;
    float*    h1    = ws + o; o += (size_t)N * H;
    float*    h2    = ws + o; o += (size_t)N * H;   // agg1, then relu(agg1+b1) in place
    float*    as    = ws + o; o += (size_t)N;
    float*    ad    = ws + o; o += (size_t)N;
    float*    denom = ws + o; o += (size_t)N;
    unsigned* menc  = (unsigned*)(ws + o); o += (size_t)N;
    float*    p     = ws + o; o += (size_t)ET;
    float*    g     = ws + o; o += (size_t)N * C;
    float*    agg2  = ws + o; o += (size_t)N * C;

    auto cdiv = [](long long a, long long b) { return (unsigned)((a + b - 1) / b); };
    dim3 blk(WG);

    // ================= layer 1 =================
    zero_f32<<<cdiv((long long)N * H, WG), blk, 0, stream>>>(h2, (long long)N * H);
    zero_f32<<<cdiv(N, WG), blk, 0, stream>>>(denom, N);
    fill_u32<<<cdiv(N, WG), blk, 0, stream>>>(menc, ENC_NEG_INF, N);

    const int mtiles = (N + 15) / 16;            // 4 n-tiles, 4 waves/block
    gemm1_wmma<<<dim3(mtiles), dim3(128), 0, stream>>>(x, W1, h1, N);

    alpha_kernel<<<cdiv(N, WG), blk, 0, stream>>>(h1, a_s1, a_d1, as, ad, N, H);
    edge_max_kernel<<<cdiv(ET, WG), blk, 0, stream>>>(ei, E, N, as, ad, menc);
    edge_p_kernel<<<cdiv(ET, WG), blk, 0, stream>>>(ei, E, N, as, ad, menc, p, denom);
    edge_agg64_kernel<<<cdiv(ET * 4, WG), blk, 0, stream>>>(ei, E, N, p, denom, h1, h2);
    bias_relu_kernel<<<cdiv((long long)N * H, WG), blk, 0, stream>>>(h2, b1, (long long)N * H, H - 1);

    // ================= layer 2 =================
    zero_f32<<<cdiv((long long)N * C, WG), blk, 0, stream>>>(agg2, (long long)N * C);
    zero_f32<<<cdiv(N, WG), blk, 0, stream>>>(denom, N);
    fill_u32<<<cdiv(N, WG), blk, 0, stream>>>(menc, ENC_NEG_INF, N);

    proj2_kernel<<<cdiv(N, WG), blk, 0, stream>>>(h2, W2, a_s2, a_d2, g, as, ad, N);
    edge_max_kernel<<<cdiv(ET, WG), blk, 0, stream>>>(ei, E, N, as, ad, menc);
    edge_p_kernel<<<cdiv(ET, WG), blk, 0, stream>>>(ei, E, N, as, ad, menc, p, denom);
    edge_agg2_kernel<<<cdiv(ET, WG), blk, 0, stream>>>(ei, E, N, p, denom, g, agg2);
    bias_out_kernel<<<cdiv((long long)N * C, WG), blk, 0, stream>>>(agg2, b2, out, (long long)N * C, C - 1);
}